// KNNC_20272245637217
// MI455X (gfx1250) — compile-verified
//
#include <hip/hip_runtime.h>
#include <cstdint>
#include <cstddef>

// KNN classify: per row of 8192x16384 f32 distances, top-16 smallest -> labels -> mode.
// One wave32 per row; 4-deep async global->LDS streaming pipeline (CDNA5 path).

#define N_COLS        16384
#define TILE_FLOATS   512
#define TILE_BYTES    2048
#define TILES         (N_COLS / TILE_FLOATS)   // 32
#define NBUF          4
#define WAVES_PER_BLK 4
#define BLOCK_THREADS (WAVES_PER_BLK * 32)

__device__ __forceinline__ int shfl_i(int v, int src) {
  return __builtin_amdgcn_ds_bpermute(src << 2, v);
}
__device__ __forceinline__ float shfl_f(float v, int src) {
  return __int_as_float(shfl_i(__float_as_int(v), src));
}

// Wave-uniform 64-bit value -> SGPR pair (legal for "s" asm constraint).
__device__ __forceinline__ uint64_t uniform64(uint64_t v) {
  uint32_t lo = __builtin_amdgcn_readfirstlane((uint32_t)v);
  uint32_t hi = __builtin_amdgcn_readfirstlane((uint32_t)(v >> 32));
  return ((uint64_t)hi << 32) | lo;
}

// CDNA5 async copy, GVS form: copies 16 bytes from global address saddr + vglob to the LDS address in vlds; tracked by ASYNCcnt.
#define ASYNC_LDSB128(vlds_, vglob_, sbase_)                                 \
  asm volatile("global_load_async_to_lds_b128 %0, %1, %2"                    \
               :: "v"(vlds_), "v"(vglob_), "s"(sbase_) : "memory")

#define WAIT_ASYNC(n) asm volatile("s_wait_asynccnt " #n ::: "memory")

__global__ __launch_bounds__(BLOCK_THREADS) void knn_mode_kernel(
    const float* __restrict__ dist, const int* __restrict__ labels,
    float* __restrict__ out, int rows) {
  __shared__ float smem[WAVES_PER_BLK * NBUF * TILE_FLOATS];  // 32 KB

  const int lane = threadIdx.x & 31;
  const int wave = threadIdx.x >> 5;
  const int row  = blockIdx.x * WAVES_PER_BLK + wave;
  if (row >= rows) return;  // wave-uniform exit; no block barriers used

  const uint32_t ldsWaveBase =
      (uint32_t)(uintptr_t)smem + (uint32_t)wave * (NBUF * TILE_BYTES);
  const uint64_t gbase =
      uniform64((uint64_t)(uintptr_t)(dist + (size_t)row * N_COLS));

  // Loop-invariant per-lane address registers.
  uint32_t vglob[4];            // global offset within a tile
  uint32_t vlds[NBUF][4];       // LDS destination addresses
#pragma unroll
  for (int k = 0; k < 4; ++k)
    vglob[k] = (uint32_t)lane * 16u + (uint32_t)k * 512u;
#pragma unroll
  for (int b = 0; b < NBUF; ++b)
#pragma unroll
    for (int k = 0; k < 4; ++k)
      vlds[b][k] = ldsWaveBase + (uint32_t)b * TILE_BYTES +
                   (uint32_t)lane * 16u + (uint32_t)k * 512u;

  // Prologue: fill all 4 buffers (tiles 0..3) -> 16 outstanding async ops.
#pragma unroll
  for (int t = 0; t < NBUF; ++t) {
    uint64_t sb = gbase + (uint64_t)t * TILE_BYTES;
#pragma unroll
    for (int k = 0; k < 4; ++k) ASYNC_LDSB128(vlds[t][k], vglob[k], sb);
  }

  const float INF = __int_as_float(0x7f800000);
  float val  = INF;  // lanes 0..15: sorted (ascending) top-16 values
  int   idxv = 0;    // matching column indices
  float t16  = INF;  // wave-uniform 16th-smallest (lane 15's val)

  auto process = [&](int tile, int buf_c) __attribute__((always_inline)) {
    const float* lbuf = smem + (size_t)wave * (NBUF * TILE_FLOATS) +
                        (size_t)buf_c * TILE_FLOATS;
#pragma unroll
    for (int k = 0; k < 4; ++k) {
      float4 x = *(const float4*)(lbuf + k * 128 + lane * 4);
      float mn = fminf(fminf(x.x, x.y), fminf(x.z, x.w));
      if (__ballot(mn < t16)) {  // rare after warm-up
        float xs[4] = {x.x, x.y, x.z, x.w};
        const int ebase = tile * TILE_FLOATS + k * 128;
#pragma unroll 1
        for (int j = 0; j < 4; ++j) {
          unsigned mask = (unsigned)__ballot(xs[j] < t16);
          while (mask) {  // wave-uniform loop
            int src = __ffs(mask) - 1;
            mask &= mask - 1;
            float v = shfl_f(xs[j], src);     // broadcast candidate
            if (v < t16) {                    // re-check vs. updated threshold
              int col = ebase + src * 4 + j;
              int pos = __popc((unsigned)__ballot(val < v));  // <= 15
              float pv = shfl_f(val, lane - 1);
              int   pi = shfl_i(idxv, lane - 1);
              float nv = (lane < pos) ? val  : ((lane == pos) ? v   : pv);
              int   ni = (lane < pos) ? idxv : ((lane == pos) ? col : pi);
              val  = (lane < 16) ? nv : INF;  // keep upper lanes inert
              idxv = (lane < 16) ? ni : 0;
              t16  = shfl_f(val, 15);
            }
          }
        }
      }
    }
  };

  // Steady state: process tile t with tiles t+1..t+3 in flight; refill buffer.
  for (int tb = 0; tb < TILES - NBUF; tb += NBUF) {
#pragma unroll
    for (int b = 0; b < NBUF; ++b) {
      WAIT_ASYNC(12);                  // oldest 4 (this tile) completed
      process(tb + b, b);
      uint64_t sb = gbase + (uint64_t)(tb + b + NBUF) * TILE_BYTES;
#pragma unroll
      for (int k = 0; k < 4; ++k) ASYNC_LDSB128(vlds[b][k], vglob[k], sb);
    }
  }
  // Tail: drain tiles TILES-4..TILES-1 with shrinking waits.
#pragma unroll
  for (int b = 0; b < NBUF; ++b) {
    if      (b == 0) WAIT_ASYNC(12);
    else if (b == 1) WAIT_ASYNC(8);
    else if (b == 2) WAIT_ASYNC(4);
    else             WAIT_ASYNC(0);
    process(TILES - NBUF + b, b);
  }

  // ---- labels + majority vote (ties -> smallest class), lanes 0..15 ----
  int lab = labels[(lane < 16) ? idxv : 0];
  int cnt = 1;
#pragma unroll 1
  for (int s = 1; s < 16; ++s) {
    int o = shfl_i(lab, (lane + s) & 15);
    cnt += (o == lab) ? 1 : 0;
  }
  int key = (cnt << 10) | (1023 - lab);  // max count, then min label
#pragma unroll
  for (int off = 8; off >= 1; off >>= 1) {
    int ok = shfl_i(key, lane ^ off);
    key = (key > ok) ? key : ok;
  }
  if (lane == 0) out[row] = (float)(1023 - (key & 1023));
}

extern "C" void kernel_launch(void* const* d_in, const int* in_sizes, int n_in,
                              void* d_out, int out_size, void* d_ws, size_t ws_size,
                              hipStream_t stream) {
  (void)n_in; (void)d_ws; (void)ws_size; (void)out_size;
  const float* dist   = (const float*)d_in[0];
  const int*   labels = (const int*)d_in[1];
  float*       out    = (float*)d_out;
  int rows = in_sizes[0] / N_COLS;  // 8192
  dim3 grid((rows + WAVES_PER_BLK - 1) / WAVES_PER_BLK), block(BLOCK_THREADS);
  hipLaunchKernelGGL(knn_mode_kernel, grid, block, 0, stream, dist, labels, out, rows);
}